// Output_29162827940664
// MI455X (gfx1250) — compile-verified
//
#include <hip/hip_runtime.h>

// MLP head: only the last time-step of x survives to the output, so we do
// tiny 32x512x512 and 32x512x256 f32 WMMA GEMMs + softmax instead of the
// reference's full-T GEMMs (4096x less work, ~2MB total traffic).

typedef float v2f __attribute__((ext_vector_type(2)));
typedef float v8f __attribute__((ext_vector_type(8)));

#define NB    32      // batch
#define CIN   512
#define CHID  512
#define COUT  256
#define TLEN  4096

// -------- 1) gather x[:, :, T-1] into a dense 32x512 f32 matrix ------------
__global__ void gather_last(const float* __restrict__ x, float* __restrict__ xlast) {
    int idx = blockIdx.x * blockDim.x + threadIdx.x;   // 0 .. 32*512-1
    // x is (B, C_IN, T) contiguous: element (b,c,T-1) at flat (b*CIN+c)*T + T-1
    xlast[idx] = x[(size_t)idx * TLEN + (TLEN - 1)];
}

// -------- 2/3) D(16x16 tile of MxN) = A(MxK) * W(NxK)^T + bias[, relu] -----
// One wave32 per 16x16 output tile; K-loop of V_WMMA_F32_16X16X4_F32.
// A-frag (16x4, ISA 7.12.2): lane m = lane&15, k = (lane>>4)*2 + {0,1}
// B-frag (4x16):             lane n = lane&15, k = (lane>>4)*2 + {0,1}
// C/D   (16x16, 8 VGPRs):    row = (lane>>4)*8 + r, col = lane&15
template<int K, bool RELU>
__global__ void wmma_gemm_bias(const float* __restrict__ A,     // M x K row-major
                               const float* __restrict__ W,     // N x K row-major
                               const float* __restrict__ bias,  // N
                               float* __restrict__ D, int ldd) {
    const int ntile = blockIdx.x;
    const int mtile = blockIdx.y;
    const int lane  = threadIdx.x;      // 0..31, all lanes active (EXEC==~0)
    const int half  = lane >> 4;
    const int l15   = lane & 15;
    const int kb    = half * 2;

    const float* arow = A + (size_t)(mtile * 16 + l15) * K;  // A row m=l15
    const float* wrow = W + (size_t)(ntile * 16 + l15) * K;  // W row n=l15 (B col)

    // Accumulator pre-loaded with bias: bias depends only on col = l15,
    // identical across all 8 C/D VGPRs of a lane.
    const float bval = bias[ntile * 16 + l15];
    v8f c;
    #pragma unroll
    for (int r = 0; r < 8; ++r) c[r] = bval;

    #pragma unroll 8
    for (int k0 = 0; k0 < K; k0 += 4) {
        v2f a, b;
        a.x = arow[k0 + kb];
        a.y = arow[k0 + kb + 1];
        b.x = wrow[k0 + kb];
        b.y = wrow[k0 + kb + 1];
        // (neg_a, A, neg_b, B, c_mod, C, reuse_a, reuse_b)
        c = __builtin_amdgcn_wmma_f32_16x16x4_f32(false, a, false, b,
                                                  (short)0, c, false, false);
    }

    #pragma unroll
    for (int r = 0; r < 8; ++r) {
        float v = c[r];
        if (RELU) v = fmaxf(v, 0.0f);
        D[(size_t)(mtile * 16 + half * 8 + r) * ldd + ntile * 16 + l15] = v;
    }
}

// -------- 4) row softmax over 256 columns ----------------------------------
__global__ void softmax_rows(const float* __restrict__ Y, float* __restrict__ out) {
    __shared__ float red[COUT];
    const int row = blockIdx.x;
    const int tid = threadIdx.x;            // 0..255
    const float v = Y[row * COUT + tid];

    red[tid] = v;
    __syncthreads();
    #pragma unroll
    for (int s = COUT / 2; s > 0; s >>= 1) {
        if (tid < s) red[tid] = fmaxf(red[tid], red[tid + s]);
        __syncthreads();
    }
    const float m = red[0];
    __syncthreads();

    const float e = expf(v - m);
    red[tid] = e;
    __syncthreads();
    #pragma unroll
    for (int s = COUT / 2; s > 0; s >>= 1) {
        if (tid < s) red[tid] += red[tid + s];
        __syncthreads();
    }
    out[row * COUT + tid] = e / red[0];
}

extern "C" void kernel_launch(void* const* d_in, const int* in_sizes, int n_in,
                              void* d_out, int out_size, void* d_ws, size_t ws_size,
                              hipStream_t stream) {
    const float* x  = (const float*)d_in[0];   // (32, 512, 4096)
    const float* W1 = (const float*)d_in[1];   // (512, 512)
    const float* b1 = (const float*)d_in[2];   // (512,)
    const float* W2 = (const float*)d_in[3];   // (256, 512)
    const float* b2 = (const float*)d_in[4];   // (256,)
    float* out = (float*)d_out;                // (32, 256)

    float* xlast = (float*)d_ws;               // 32*512 f32 =  64 KB
    float* h     = xlast + NB * CIN;           // 32*512 f32 =  64 KB
    float* y     = h     + NB * CHID;          // 32*256 f32 =  32 KB

    gather_last<<<(NB * CIN) / 256, 256, 0, stream>>>(x, xlast);

    wmma_gemm_bias<CIN, true>
        <<<dim3(CHID / 16, NB / 16), 32, 0, stream>>>(xlast, W1, b1, h, CHID);

    wmma_gemm_bias<CHID, false>
        <<<dim3(COUT / 16, NB / 16), 32, 0, stream>>>(h, W2, b2, y, COUT);

    softmax_rows<<<NB, COUT, 0, stream>>>(y, out);
}